// FrameSwitch_17841294148082
// MI455X (gfx1250) — compile-verified
//
#include <hip/hip_runtime.h>

#ifndef __has_builtin
#define __has_builtin(x) 0
#endif

#define P_SWAP 0.5f
#define BLOCK  256
#define CHUNK  4   // float4 (16B) transfers per thread

// gfx1250 async LDS copy path (ASYNCcnt-tracked). The builtins take
// (int4 addrspace(1)* gaddr, int4 addrspace(3)* lds, imm offset, imm cpol)
// per the hipcc diagnostic from the previous round.
#if defined(__AMDGCN__) &&                                          \
    __has_builtin(__builtin_amdgcn_global_load_async_to_lds_b128) && \
    __has_builtin(__builtin_amdgcn_global_store_async_from_lds_b128) && \
    __has_builtin(__builtin_amdgcn_s_wait_asynccnt)
#define USE_ASYNC_LDS 1
typedef int v4i __attribute__((ext_vector_type(4)));
typedef __attribute__((address_space(1))) v4i* gas_b128;   // global (AS1) int4*
typedef __attribute__((address_space(3))) v4i* las_b128;   // LDS    (AS3) int4*
#else
#define USE_ASYNC_LDS 0
#endif

// ---------------------------------------------------------------------------
// Kernel 1: resolve the serial scan into a per-frame gather index src[i] such
// that out[i] == video[src[i]]. 128 sequential steps on LDS-staged flags.
// ---------------------------------------------------------------------------
__global__ void fs_resolve_src(const float* __restrict__ coin,
                               const int*   __restrict__ nbr,
                               int*         __restrict__ src,
                               int N) {
    __shared__ int doS[1024];
    __shared__ int nbS[1024];
    __shared__ int srcS[1024];
    const int i = threadIdx.x;
    if (i < N) {                       // parallel load: one memory latency total
        doS[i] = (coin[i] < P_SWAP) ? 1 : 0;
        nbS[i] = nbr[i];
    }
    __syncthreads();
    if (i == 0) {                      // 128-step scan in LDS/registers
        int prev = 0;
        for (int k = 0; k < N; ++k) {
            int s = k;
            if (doS[k]) {
                if (k == 0)                         s = (N > 1) ? 1 : 0; // orig next
                else if (k == N - 1 || nbS[k] == 0) s = prev;            // carry
                else                                s = k + 1;           // orig next
            }
            srcS[k] = s;
            prev = s;
        }
    }
    __syncthreads();
    if (i < N) src[i] = srcS[i];
}

// ---------------------------------------------------------------------------
// Kernel 2: out[frame] = video[src[frame]] — HBM-bound streaming gather-copy.
// Each block moves BLOCK*CHUNK float4 (16 KB) of one frame. Async path stages
// through LDS with GLOBAL_LOAD_ASYNC_TO_LDS_B128 / STORE_ASYNC_FROM_LDS_B128.
// ---------------------------------------------------------------------------
__global__ __launch_bounds__(BLOCK)
void fs_gather_copy(const float4* __restrict__ video,
                    float4*       __restrict__ out,
                    const int*    __restrict__ src,
                    long long frameVecs) {
    const int       frame = blockIdx.y;
    const long long s     = (long long)src[frame];   // uniform per block
    const float4* __restrict__ inF  = video + s * frameVecs;
    float4*       __restrict__ outF = out   + (long long)frame * frameVecs;

    const long long base = (long long)blockIdx.x * (BLOCK * CHUNK);

#if USE_ASYNC_LDS
    __shared__ float4 stage[BLOCK * CHUNK];          // 16 KB per block
    #pragma unroll
    for (int c = 0; c < CHUNK; ++c) {
        const long long idx = base + (long long)c * BLOCK + threadIdx.x;
        if (idx < frameVecs) {
            __builtin_amdgcn_global_load_async_to_lds_b128(
                (gas_b128)(v4i*)(void*)(inF + idx),
                (las_b128)(v4i*)(void*)&stage[c * BLOCK + threadIdx.x],
                /*offset=*/0, /*cpol=*/0);
        }
    }
    __builtin_amdgcn_s_wait_asynccnt(0);             // LDS slots filled (per-wave)
    #pragma unroll
    for (int c = 0; c < CHUNK; ++c) {
        const long long idx = base + (long long)c * BLOCK + threadIdx.x;
        if (idx < frameVecs) {
            __builtin_amdgcn_global_store_async_from_lds_b128(
                (gas_b128)(v4i*)(void*)(outF + idx),
                (las_b128)(v4i*)(void*)&stage[c * BLOCK + threadIdx.x],
                /*offset=*/0, /*cpol=*/0);
        }
    }
    __builtin_amdgcn_s_wait_asynccnt(0);
#else
    #pragma unroll
    for (int c = 0; c < CHUNK; ++c) {
        const long long idx = base + (long long)c * BLOCK + threadIdx.x;
        if (idx < frameVecs) outF[idx] = inF[idx];   // global_load/store_b128
    }
#endif
}

// ---------------------------------------------------------------------------
// Inputs (setup_inputs order): video f32[N*3*H*W], coin f32[N], nbr i32[N].
// Output: f32[N*3*H*W]. d_ws holds src[N] (int).
// ---------------------------------------------------------------------------
extern "C" void kernel_launch(void* const* d_in, const int* in_sizes, int n_in,
                              void* d_out, int out_size, void* d_ws, size_t ws_size,
                              hipStream_t stream) {
    const float* video = (const float*)d_in[0];
    const float* coin  = (const float*)d_in[1];
    const int*   nbr   = (const int*)d_in[2];
    float*       out   = (float*)d_out;

    const int N = in_sizes[1];                                   // 128 frames
    const long long frameElems = (long long)in_sizes[0] / (N > 0 ? N : 1);
    const long long frameVecs  = frameElems / 4;                 // 3*512*512 % 4 == 0
    int* src = (int*)d_ws;

    fs_resolve_src<<<dim3(1), dim3(N), 0, stream>>>(coin, nbr, src, N);

    const long long perBlock = (long long)BLOCK * CHUNK;
    const int gridX = (int)((frameVecs + perBlock - 1) / perBlock);  // 192 for 512x512x3
    dim3 grid(gridX, N, 1);
    fs_gather_copy<<<grid, dim3(BLOCK), 0, stream>>>(
        (const float4*)video, (float4*)out, src, frameVecs);
}